// LigerJSD_69707319214415
// MI455X (gfx1250) — compile-verified
//
#include <hip/hip_runtime.h>
#include <hip/hip_bf16.h>
#include <stdint.h>

#define THREADS 256
#define STAGES  4   // async pipeline depth (power of two)

// ---------------------------------------------------------------------------
// JSD(beta=0.5) pointwise term for log-prob inputs p, q:
//   P = e^p, Q = e^q, m = 0.5*(P+Q)
//   loss = 0.5*(P*p + Q*q) - m*log(m)   (algebraically == Liger generalized JSD)
// v_exp_f32 / v_log_f32 via fast intrinsics; inputs are log-softmax values
// (~[-16, 0] for N(0,1) logits over a 32000 vocab) so exp/log stay in range.
// ---------------------------------------------------------------------------
__device__ __forceinline__ float jsd_term(float pl, float ql) {
    float P  = __expf(pl);
    float Q  = __expf(ql);
    float m  = 0.5f * (P + Q);
    float lm = __logf(m);
    return 0.5f * (P * (pl - lm) + Q * (ql - lm));
}

__device__ __forceinline__ float jsd4(const float4& pv, const float4& qv) {
    return jsd_term(pv.x, qv.x) + jsd_term(pv.y, qv.y) +
           jsd_term(pv.z, qv.z) + jsd_term(pv.w, qv.w);
}

// gfx1250 async global->LDS copy, GV mode (64-bit vaddr), 16B per lane.
// vdst = 32-bit LDS byte offset VGPR, vaddr = 64-bit global address pair.
#define ASYNC_LOAD_B128(ldsoff, gaddr)                                        \
    asm volatile("global_load_async_to_lds_b128 %0, %1, off"                  \
                 :: "v"(ldsoff), "v"(gaddr) : "memory")

// ---------------------------------------------------------------------------
// Kernel 1: streaming partial reduction.
// Each wave runs a private 4-stage async pipeline: 2 async B128 per stage
// (p-stream + q-stream), 8 outstanding async ops per wave. Each lane consumes
// only the 16 bytes it staged itself -> no cross-lane LDS hazards in the hot
// loop; sync is purely s_wait_asynccnt / s_wait_dscnt (no barriers).
// The steady-state loop is unrolled by STAGES so the slot index is a literal:
// ds_load_b128 uses immediate offsets and no register-array selects remain.
// ---------------------------------------------------------------------------
extern "C" __global__ void __launch_bounds__(THREADS)
jsd_partial_kernel(const float* __restrict__ p, const float* __restrict__ q,
                   float* __restrict__ partial,
                   size_t total4, size_t stride4, size_t iters)
{
    __shared__ float4 sp[STAGES][THREADS];
    __shared__ float4 sq[STAGES][THREADS];
    __shared__ float  red[THREADS];

    const int    tid  = threadIdx.x;
    const size_t idx0 = (size_t)blockIdx.x * THREADS + tid;

    // Per-lane LDS byte offsets (flat low 32 bits == LDS offset), invariant,
    // one VGPR each since the stage index below is always literal.
    unsigned ldsP[STAGES], ldsQ[STAGES];
    #pragma unroll
    for (int s = 0; s < STAGES; ++s) {
        ldsP[s] = (unsigned)(size_t)(&sp[s][tid]);
        ldsQ[s] = (unsigned)(size_t)(&sq[s][tid]);
    }

    float acc = 0.0f;

    if (iters >= (size_t)STAGES) {
        // Prologue: fill the pipeline (8 async B128 in flight per wave).
        #pragma unroll
        for (int s = 0; s < STAGES; ++s) {
            size_t g = idx0 + (size_t)s * stride4;
            ASYNC_LOAD_B128(ldsP[s], (unsigned long long)(uintptr_t)(p + 4 * g));
            ASYNC_LOAD_B128(ldsQ[s], (unsigned long long)(uintptr_t)(q + 4 * g));
        }

        // Steady state, unrolled by STAGES so `slot` is a compile-time
        // constant in every stage (constant LDS offsets, no cselect chains).
        const size_t groups = iters / STAGES;
        for (size_t gi = 0; gi < groups; ++gi) {
            #pragma unroll
            for (int s = 0; s < STAGES; ++s) {
                const size_t it = gi * STAGES + (size_t)s;

                // Oldest stage's 2 loads done (loads complete in order);
                // keep (STAGES-1)*2 = 6 younger ops in flight.
                asm volatile("s_wait_asynccnt 0x6" ::: "memory");

                float4 pv = sp[s][tid];   // ds_load_b128, imm offset
                float4 qv = sq[s][tid];   // ds_load_b128, imm offset

                // WAR fence: LDS reads landed in VGPRs before the slot is
                // rewritten (LDS ops vs async writes are unordered).
                asm volatile("s_wait_dscnt 0x0" ::: "memory");

                const size_t nxt = it + STAGES;
                if (nxt < iters) {   // uniform scalar branch
                    size_t g = idx0 + nxt * stride4;
                    ASYNC_LOAD_B128(ldsP[s], (unsigned long long)(uintptr_t)(p + 4 * g));
                    ASYNC_LOAD_B128(ldsQ[s], (unsigned long long)(uintptr_t)(q + 4 * g));
                }

                // Compute after the refill is in flight.
                acc += jsd4(pv, qv);
            }
        }

        // Leftover stages (iters % STAGES), still pipelined.
        for (size_t it = groups * STAGES; it < iters; ++it) {
            const int slot = (int)(it & (STAGES - 1));
            asm volatile("s_wait_asynccnt 0x0" ::: "memory");
            float4 pv = sp[slot][tid];
            float4 qv = sq[slot][tid];
            acc += jsd4(pv, qv);
            asm volatile("s_wait_dscnt 0x0" ::: "memory");
        }
        asm volatile("s_wait_asynccnt 0x0" ::: "memory");
    } else {
        // Degenerate tiny-input path: no pipelining, full wait each stage.
        for (size_t it = 0; it < iters; ++it) {
            const int slot = (int)(it & (STAGES - 1));
            size_t g = idx0 + it * stride4;
            ASYNC_LOAD_B128(ldsP[slot], (unsigned long long)(uintptr_t)(p + 4 * g));
            ASYNC_LOAD_B128(ldsQ[slot], (unsigned long long)(uintptr_t)(q + 4 * g));
            asm volatile("s_wait_asynccnt 0x0" ::: "memory");
            float4 pv = sp[slot][tid];
            float4 qv = sq[slot][tid];
            acc += jsd4(pv, qv);
            asm volatile("s_wait_dscnt 0x0" ::: "memory");
        }
    }

    // Generic tail (empty for 4096x32000 with 2000 blocks): direct loads.
    for (size_t g = idx0 + iters * stride4; g < total4; g += stride4) {
        float4 pv = *(const float4*)(p + 4 * g);
        float4 qv = *(const float4*)(q + 4 * g);
        acc += jsd4(pv, qv);
    }

    // Deterministic in-block tree reduction.
    red[tid] = acc;
    __syncthreads();
    #pragma unroll
    for (int off = THREADS / 2; off > 0; off >>= 1) {
        if (tid < off) red[tid] += red[tid + off];
        __syncthreads();
    }
    if (tid == 0) partial[blockIdx.x] = red[0];
}

// ---------------------------------------------------------------------------
// Kernel 2: deterministic final reduction of per-block partials (+ scalar
// remainder if total % 4 != 0), scaled by 1/n_rows.
// ---------------------------------------------------------------------------
extern "C" __global__ void __launch_bounds__(THREADS)
jsd_final_kernel(const float* __restrict__ partial, int nparts,
                 const float* __restrict__ p, const float* __restrict__ q,
                 size_t total, size_t total4, float inv_rows,
                 float* __restrict__ out)
{
    __shared__ float red[THREADS];
    const int tid = threadIdx.x;

    float acc = 0.0f;
    for (int i = tid; i < nparts; i += THREADS) acc += partial[i];  // fixed order
    if (tid == 0) {
        for (size_t i = total4 * 4; i < total; ++i) acc += jsd_term(p[i], q[i]);
    }

    red[tid] = acc;
    __syncthreads();
    #pragma unroll
    for (int off = THREADS / 2; off > 0; off >>= 1) {
        if (tid < off) red[tid] += red[tid + off];
        __syncthreads();
    }
    if (tid == 0) out[0] = red[0] * inv_rows;
}

// ---------------------------------------------------------------------------
// Host launcher. No atomics, no allocation, graph-capture safe, deterministic.
// ---------------------------------------------------------------------------
extern "C" void kernel_launch(void* const* d_in, const int* in_sizes, int n_in,
                              void* d_out, int out_size, void* d_ws, size_t ws_size,
                              hipStream_t stream) {
    (void)n_in; (void)out_size;
    const float* p   = (const float*)d_in[0];
    const float* q   = (const float*)d_in[1];
    float*       out = (float*)d_out;
    float*       ws  = (float*)d_ws;

    const size_t total  = (size_t)in_sizes[0];   // B*V = 4096*32000
    const size_t total4 = total / 4;
    const int    rows   = (int)(total / 32000);  // V = 32000 per reference
    const float  inv_rows = 1.0f / (float)(rows > 0 ? rows : 1);

    // 2000 blocks * 256 threads -> exactly 64 pipelined chunks/thread for this
    // shape (32,768,000 float4 pairs), zero tail. 16k waves keep ~64 MB of
    // async traffic in flight, well above the ~25 MB BW-delay product.
    int blocks = 2000;
    if (ws_size < (size_t)blocks * sizeof(float))
        blocks = (int)(ws_size / sizeof(float));
    if (blocks < 1) blocks = 1;

    const size_t stride4 = (size_t)blocks * THREADS;  // float4 grid stride
    const size_t iters   = total4 / stride4;          // host-side: no u64 div on GPU

    jsd_partial_kernel<<<blocks, THREADS, 0, stream>>>(p, q, ws,
                                                       total4, stride4, iters);
    jsd_final_kernel<<<1, THREADS, 0, stream>>>(ws, blocks, p, q,
                                                total, total4, inv_rows, out);
}